// SVGraphConvLayer_17892833755720
// MI455X (gfx1250) — compile-verified
//
#include <hip/hip_runtime.h>
#include <hip/hip_bf16.h>

typedef float v2f __attribute__((ext_vector_type(2)));
typedef float v8f __attribute__((ext_vector_type(8)));

#define WPB 4  // 16-edge tiles (waves) per 128-thread block

__device__ __forceinline__ v8f v8zero() {
  v8f z;
#pragma unroll
  for (int i = 0; i < 8; ++i) z[i] = 0.0f;
  return z;
}

// per-wave LDS ordering (producer/consumer within the same wave)
__device__ __forceinline__ void lds_fence() {
  asm volatile("s_wait_dscnt 0" ::: "memory");
}

// fp32 WMMA: D(16x16,f32) = A(16x4,f32) * B(4x16,f32) + C
__device__ __forceinline__ v8f wmma4(v2f a, v2f b, v8f c) {
  return __builtin_amdgcn_wmma_f32_16x16x4_f32(false, a, false, b, (short)0, c,
                                               false, false);
}

// Tiled GEMM: acc[nt] += A(16 x K) * B(K x NT*16).
// A fragment lane map (32-bit 16x4): lanes 0-15 -> M, VGPR pair -> K, K+1 for
// low half, K+2, K+3 for high half. B mirrored with lanes -> N.
template <int NT, typename FA, typename FB>
__device__ __forceinline__ void gemm16(v8f* acc, int K, FA a_ld, FB b_ld) {
  const int lane = threadIdx.x & 31;
  const int mn = lane & 15;
  const int kh = (lane >> 4) << 1;
  for (int k0 = 0; k0 < K; k0 += 4) {
    v2f a;
    a.x = a_ld(mn, k0 + kh);
    a.y = a_ld(mn, k0 + kh + 1);
#pragma unroll
    for (int nt = 0; nt < NT; ++nt) {
      v2f b;
      b.x = b_ld(k0 + kh, nt * 16 + mn);
      b.y = b_ld(k0 + kh + 1, nt * 16 + mn);
      acc[nt] = wmma4(a, b, acc[nt]);
    }
  }
}

__device__ __forceinline__ void atomicMaxF(float* addr, float val) {
  if (val >= 0.0f)
    atomicMax((int*)addr, __float_as_int(val));
  else
    atomicMin((unsigned int*)addr, (unsigned int)__float_as_int(val));
}

struct SvpW { const float *Wh, *Ws, *bs, *Wmu, *Wg, *bg; };

struct Params {
  const float *xs, *xv, *es_in, *ev_in, *rot;
  const int *esrc, *edst;
  SvpW tgt, srcw, edg, o1, o2;
  const float *v2s, *s2v;
  // workspace
  float *ws_srs, *ws_srv, *ws_es, *ws_evv, *ws_vfs, *ws_sfv, *ws_lg;
  float *nmax, *nden, *cnt, *ssum, *vsum;
  float* out;
  int N, E;
};

// GVP scalar-vector perceptron on a 16-edge tile, out_s=64, out_v=16.
// s_ld(m,k): scalar input; v_ld(m,c,j): vector input (local frame).
// scr: per-wave LDS scratch, >= 16*64 floats.
// sO[4]: s_out frags (D layout, N-tile t = channels 16t..16t+15 = head t).
// vO[3]: v_out frags per spatial component j.
template <int IN_S, int IN_V, int H, typename FS, typename FV>
__device__ void svp_tile(const SvpW& w, FS s_ld, FV v_ld, float* scr,
                         v8f sO[4], v8f vO[3]) {
  constexpr int HT = H / 16;
  const int lane = threadIdx.x & 31;
  const int nn = lane & 15;
  const int hi = lane >> 4;
  v8f n2[HT];
#pragma unroll
  for (int t = 0; t < HT; ++t) n2[t] = v8zero();
  v8f vmu[3];
  for (int j = 0; j < 3; ++j) {
    v8f vh[HT];
#pragma unroll
    for (int t = 0; t < HT; ++t) vh[t] = v8zero();
    gemm16<HT>(vh, IN_V,
               [&](int m, int k) { return v_ld(m, k, j); },
               [&](int k, int n) { return w.Wh[n * IN_V + k]; });
    lds_fence();  // WAR vs previous iter's reads of scr
#pragma unroll
    for (int t = 0; t < HT; ++t)
#pragma unroll
      for (int r = 0; r < 8; ++r)
        scr[(r + 8 * hi) * H + t * 16 + nn] = vh[t][r];
    lds_fence();
    vmu[j] = v8zero();
    gemm16<1>(&vmu[j], H,
              [&](int m, int k) { return scr[m * H + k]; },
              [&](int k, int n) { return w.Wmu[n * H + k]; });
#pragma unroll
    for (int t = 0; t < HT; ++t) n2[t] += vh[t] * vh[t];
  }
  // norms -> scratch (A layout source for the Ws GEMM tail)
  lds_fence();
#pragma unroll
  for (int t = 0; t < HT; ++t)
#pragma unroll
    for (int r = 0; r < 8; ++r)
      scr[(r + 8 * hi) * H + t * 16 + nn] = sqrtf(n2[t][r] + 1e-8f);
  lds_fence();
#pragma unroll
  for (int t = 0; t < 4; ++t) sO[t] = v8zero();
  gemm16<4>(sO, IN_S + H,
            [&](int m, int k) {
              return (k < IN_S) ? s_ld(m, k) : scr[m * H + (k - IN_S)];
            },
            [&](int k, int n) { return w.Ws[k * 64 + n]; });
#pragma unroll
  for (int t = 0; t < 4; ++t) {
    float b = w.bs[t * 16 + nn];
#pragma unroll
    for (int r = 0; r < 8; ++r) sO[t][r] = fmaxf(sO[t][r] + b, 0.0f);
  }
  // s_out -> scratch for the gate GEMM
  lds_fence();
#pragma unroll
  for (int t = 0; t < 4; ++t)
#pragma unroll
    for (int r = 0; r < 8; ++r)
      scr[(r + 8 * hi) * 64 + t * 16 + nn] = sO[t][r];
  lds_fence();
  v8f g = v8zero();
  gemm16<1>(&g, 64,
            [&](int m, int k) { return scr[m * 64 + k]; },
            [&](int k, int n) { return w.Wg[k * 16 + n]; });
  {
    float b = w.bg[nn];
#pragma unroll
    for (int r = 0; r < 8; ++r) g[r] = 1.0f / (1.0f + __expf(-(g[r] + b)));
  }
#pragma unroll
  for (int j = 0; j < 3; ++j) vO[j] = vmu[j] * g;
}

// ---------------- kernel 0: init node accumulators -------------------------
__global__ void k_init(Params P) {
  int i = blockIdx.x * blockDim.x + threadIdx.x;
  if (i < P.N * 16) { P.nmax[i] = -3.0e38f; P.nden[i] = 0.0f; }
  if (i < P.N) P.cnt[i] = 0.0f;
  if (i < P.N * 64) P.ssum[i] = 0.0f;
  if (i < P.N * 48) P.vsum[i] = 0.0f;
}

// ---------------- kernel 1: edge front (3 SVPs + logits) -------------------
__global__ __launch_bounds__(32 * WPB) void k_front(Params P) {
  const int wv = threadIdx.x >> 5;
  const int lane = threadIdx.x & 31;
  const int tile = blockIdx.x * WPB + wv;
  const int ntiles = (P.E + 15) >> 4;
  __shared__ float s_vi[WPB][16 * 16 * 3];
  __shared__ float s_vj[WPB][16 * 16 * 3];
  __shared__ float s_ev[WPB][16 * 8 * 3];
  __shared__ float s_scr[WPB][16 * 64];
  __shared__ int s_idx[WPB][32];  // [m*2] = src node, [m*2+1] = dst node
  if (tile >= ntiles) return;
  const int e0 = tile << 4;
  float* vi = s_vi[wv];
  float* vj = s_vj[wv];
  float* ev = s_ev[wv];
  float* scr = s_scr[wv];
  int* idx = s_idx[wv];
  const int nn = lane & 15, hi = lane >> 4;
  if (lane < 16) {
    int e = e0 + lane;
    if (e >= P.E) e = P.E - 1;
    idx[lane * 2 + 0] = P.esrc[e];
    idx[lane * 2 + 1] = P.edst[e];
  }
  lds_fence();
  // gather + rotate v_i, v_j, edge_v into the target local frame (g2l)
  for (int t = lane; t < 640; t += 32) {
    int e_l, c;
    const float* vp;
    float* op;
    if (t < 256) {
      e_l = t >> 4; c = t & 15;
      vp = P.xv + ((size_t)idx[e_l * 2 + 1] * 16 + c) * 3;
      op = vi + e_l * 48 + c * 3;
    } else if (t < 512) {
      int tt = t - 256; e_l = tt >> 4; c = tt & 15;
      vp = P.xv + ((size_t)idx[e_l * 2 + 0] * 16 + c) * 3;
      op = vj + e_l * 48 + c * 3;
    } else {
      int tt = t - 512; e_l = tt >> 3; c = tt & 7;
      int e = e0 + e_l; if (e >= P.E) e = P.E - 1;
      vp = P.ev_in + ((size_t)e * 8 + c) * 3;
      op = ev + e_l * 24 + c * 3;
    }
    const float* R = P.rot + (size_t)idx[e_l * 2 + 1] * 9;
    float a0 = vp[0], a1 = vp[1], a2 = vp[2];
#pragma unroll
    for (int k = 0; k < 3; ++k)  // out[c,k] = sum_j v[c,j]*R[j,k]
      op[k] = a0 * R[k] + a1 * R[3 + k] + a2 * R[6 + k];
  }
  lds_fence();
  v8f tS[4], tV[3], rS[4], rV[3];
  svp_tile<64, 16, 16>(P.tgt,
      [&](int m, int k) { return P.xs[(size_t)idx[m * 2 + 1] * 64 + k]; },
      [&](int m, int k, int j) { return vi[m * 48 + k * 3 + j]; },
      scr, tS, tV);
  svp_tile<64, 16, 16>(P.srcw,
      [&](int m, int k) { return P.xs[(size_t)idx[m * 2 + 0] * 64 + k]; },
      [&](int m, int k, int j) { return vj[m * 48 + k * 3 + j]; },
      scr, rS, rV);
  v8f gS[4], gV[3];
  svp_tile<32, 8, 16>(P.edg,
      [&](int m, int k) {
        int e = e0 + m; if (e >= P.E) e = P.E - 1;
        return P.es_in[(size_t)e * 32 + k];
      },
      [&](int m, int k, int j) { return ev[m * 24 + k * 3 + j]; },
      scr, gS, gV);
  // persist e_s, sr_s, e_vv, sr_v
#pragma unroll
  for (int t = 0; t < 4; ++t)
#pragma unroll
    for (int r = 0; r < 8; ++r) {
      int e = e0 + r + 8 * hi;
      if (e < P.E) {
        P.ws_es[(size_t)e * 64 + t * 16 + nn] = gS[t][r];
        P.ws_srs[(size_t)e * 64 + t * 16 + nn] = rS[t][r];
      }
    }
#pragma unroll
  for (int j = 0; j < 3; ++j)
#pragma unroll
    for (int r = 0; r < 8; ++r) {
      int e = e0 + r + 8 * hi;
      if (e < P.E) {
        P.ws_evv[(size_t)e * 48 + nn * 3 + j] = gV[j][r];
        P.ws_srv[(size_t)e * 48 + nn * 3 + j] = rV[j][r];
      }
    }
  // stage sr_v (reuse vi) for the v2s GEMMs
  lds_fence();
#pragma unroll
  for (int j = 0; j < 3; ++j)
#pragma unroll
    for (int r = 0; r < 8; ++r)
      vi[(r + 8 * hi) * 48 + nn * 3 + j] = rV[j][r];
  lds_fence();
  // vfs = sum_j (v2s_W @ sr_v_j)^2
  v8f vfs[4];
#pragma unroll
  for (int t = 0; t < 4; ++t) vfs[t] = v8zero();
  for (int j = 0; j < 3; ++j) {
    v8f u[4];
#pragma unroll
    for (int t = 0; t < 4; ++t) u[t] = v8zero();
    gemm16<4>(u, 16,
              [&](int m, int k) { return vi[m * 48 + k * 3 + j]; },
              [&](int k, int n) { return P.v2s[n * 16 + k]; });
#pragma unroll
    for (int t = 0; t < 4; ++t) vfs[t] += u[t] * u[t];
  }
#pragma unroll
  for (int t = 0; t < 4; ++t)
#pragma unroll
    for (int r = 0; r < 8; ++r) {
      int e = e0 + r + 8 * hi;
      if (e < P.E) P.ws_vfs[(size_t)e * 64 + t * 16 + nn] = vfs[t][r];
    }
  // sfv = sr_s @ s2v_W^T  (stage sr_s in scr)
  lds_fence();
#pragma unroll
  for (int t = 0; t < 4; ++t)
#pragma unroll
    for (int r = 0; r < 8; ++r)
      scr[(r + 8 * hi) * 64 + t * 16 + nn] = rS[t][r];
  lds_fence();
  v8f sfv = v8zero();
  gemm16<1>(&sfv, 64,
            [&](int m, int k) { return scr[m * 64 + k]; },
            [&](int k, int n) { return P.s2v[n * 64 + k]; });
#pragma unroll
  for (int r = 0; r < 8; ++r) {
    int e = e0 + r + 8 * hi;
    if (e < P.E) P.ws_sfv[(size_t)e * 16 + nn] = sfv[r];
  }
  // attention logits, layout lg[e*16 + {l1(4), l2(4), l3(4), l4(4)}]
  auto dotw16 = [&](const v8f& a, const v8f& b, int q) {
#pragma unroll
    for (int r = 0; r < 8; ++r) {
      float p = a[r] * b[r];
      p += __shfl_xor(p, 1); p += __shfl_xor(p, 2);
      p += __shfl_xor(p, 4); p += __shfl_xor(p, 8);
      if (nn == 0) {
        int m = r + 8 * hi, e = e0 + m;
        if (e < P.E) {
          P.ws_lg[(size_t)e * 16 + q] = p;
          atomicMaxF(&P.nmax[(size_t)idx[m * 2 + 1] * 16 + q], p);
        }
      }
    }
  };
  auto dotw4 = [&](const v8f& pv, int qb) {
#pragma unroll
    for (int r = 0; r < 8; ++r) {
      float p = pv[r];
      p += __shfl_xor(p, 1); p += __shfl_xor(p, 2);
      if ((nn & 3) == 0) {
        int m = r + 8 * hi, e = e0 + m, q = qb + (nn >> 2);
        if (e < P.E) {
          P.ws_lg[(size_t)e * 16 + q] = p;
          atomicMaxF(&P.nmax[(size_t)idx[m * 2 + 1] * 16 + q], p);
        }
      }
    }
  };
#pragma unroll
  for (int t = 0; t < 4; ++t) dotw16(tS[t], vfs[t], t);
#pragma unroll
  for (int t = 0; t < 4; ++t) dotw16(tS[t], rS[t], 4 + t);
  {
    v8f tsum = tV[0] + tV[1] + tV[2];
    v8f pv = tsum * sfv;  // sum_j t_v[c,j]*sfv[c]
    dotw4(pv, 8);
  }
  {
    v8f pv = tV[0] * rV[0] + tV[1] * rV[1] + tV[2] * rV[2];
    dotw4(pv, 12);
  }
}

// ---------------- kernel 2: softmax exp/denominator + edge counts ----------
__global__ void k_soft(Params P) {
  int i = blockIdx.x * blockDim.x + threadIdx.x;
  if (i >= P.E * 16) return;
  int e = i >> 4, q = i & 15;
  int d = P.edst[e];
  float ex = __expf(P.ws_lg[i] - P.nmax[(size_t)d * 16 + q]);
  P.ws_lg[i] = ex;
  atomicAdd(&P.nden[(size_t)d * 16 + q], ex);
  if (q == 0) atomicAdd(&P.cnt[d], 1.0f);
}

// ---------------- kernel 3: out1+out2 SVPs, rotate back, scatter -----------
__global__ __launch_bounds__(32 * WPB) void k_back(Params P) {
  const int wv = threadIdx.x >> 5;
  const int lane = threadIdx.x & 31;
  const int tile = blockIdx.x * WPB + wv;
  const int ntiles = (P.E + 15) >> 4;
  __shared__ float s_sc[WPB][256];
  __shared__ float s_sio[WPB][16 * 64];
  __shared__ float s_vio[WPB][16 * 48];
  __shared__ float s_scr[WPB][16 * 64];
  __shared__ int s_dst[WPB][16];
  if (tile >= ntiles) return;
  const int e0 = tile << 4;
  float* sc = s_sc[wv];
  float* sio = s_sio[wv];
  float* vio = s_vio[wv];
  float* scr = s_scr[wv];
  int* dstl = s_dst[wv];
  const int nn = lane & 15, hi = lane >> 4;
  if (lane < 16) {
    int e = e0 + lane;
    if (e >= P.E) e = P.E - 1;
    dstl[lane] = P.edst[e];
  }
  lds_fence();
  for (int t = lane; t < 256; t += 32) {
    int m = t >> 4, q = t & 15;
    int e = e0 + m; if (e >= P.E) e = P.E - 1;
    sc[m * 16 + q] =
        P.ws_lg[(size_t)e * 16 + q] / (P.nden[(size_t)dstl[m] * 16 + q] + 1e-16f);
  }
  lds_fence();
  auto s_ld1 = [&](int m, int k) -> float {  // s_new = [s1, s2, e_s]
    int e = e0 + m; if (e >= P.E) e = P.E - 1;
    if (k < 64) return P.ws_vfs[(size_t)e * 64 + k] * sc[m * 16 + (k >> 4)];
    if (k < 128) {
      int kk = k - 64;
      return P.ws_srs[(size_t)e * 64 + kk] * sc[m * 16 + 4 + (kk >> 4)];
    }
    return P.ws_es[(size_t)e * 64 + (k - 128)];
  };
  auto v_ld1 = [&](int m, int c, int j) -> float {  // v_new = [v1, v2, e_vv]
    int e = e0 + m; if (e >= P.E) e = P.E - 1;
    if (c < 16) return P.ws_sfv[(size_t)e * 16 + c] * sc[m * 16 + 8 + (c >> 2)];
    if (c < 32) {
      int cc = c - 16;
      return P.ws_srv[(size_t)e * 48 + cc * 3 + j] * sc[m * 16 + 12 + (cc >> 2)];
    }
    return P.ws_evv[(size_t)e * 48 + (c - 32) * 3 + j];
  };
  v8f m1S[4], m1V[3];
  svp_tile<192, 48, 48>(P.o1, s_ld1, v_ld1, scr, m1S, m1V);
  lds_fence();
#pragma unroll
  for (int t = 0; t < 4; ++t)
#pragma unroll
    for (int r = 0; r < 8; ++r)
      sio[(r + 8 * hi) * 64 + t * 16 + nn] = m1S[t][r];
#pragma unroll
  for (int j = 0; j < 3; ++j)
#pragma unroll
    for (int r = 0; r < 8; ++r)
      vio[(r + 8 * hi) * 48 + nn * 3 + j] = m1V[j][r];
  lds_fence();
  v8f m2S[4], m2V[3];
  svp_tile<64, 16, 16>(P.o2,
      [&](int m, int k) { return sio[m * 64 + k]; },
      [&](int m, int k, int j) { return vio[m * 48 + k * 3 + j]; },
      scr, m2S, m2V);
  // scalar mean-scatter
#pragma unroll
  for (int t = 0; t < 4; ++t)
#pragma unroll
    for (int r = 0; r < 8; ++r) {
      int m = r + 8 * hi, e = e0 + m;
      if (e < P.E)
        atomicAdd(&P.ssum[(size_t)dstl[m] * 64 + t * 16 + nn], m2S[t][r]);
    }
  // vector: stage, rotate back to global frame (l2g), scatter
  lds_fence();
#pragma unroll
  for (int j = 0; j < 3; ++j)
#pragma unroll
    for (int r = 0; r < 8; ++r)
      vio[(r + 8 * hi) * 48 + nn * 3 + j] = m2V[j][r];
  lds_fence();
  for (int t = lane; t < 256; t += 32) {
    int m = t >> 4, c = t & 15;
    int e = e0 + m;
    if (e < P.E) {
      const float* R = P.rot + (size_t)dstl[m] * 9;
      float vx = vio[m * 48 + c * 3 + 0];
      float vy = vio[m * 48 + c * 3 + 1];
      float vz = vio[m * 48 + c * 3 + 2];
#pragma unroll
      for (int j = 0; j < 3; ++j) {  // out[c,j] = sum_k v[c,k]*R[j,k]
        float o = vx * R[j * 3 + 0] + vy * R[j * 3 + 1] + vz * R[j * 3 + 2];
        atomicAdd(&P.vsum[(size_t)dstl[m] * 48 + c * 3 + j], o);
      }
    }
  }
}

// ---------------- kernel 4: finalize mean, write (out_s || out_v) ----------
__global__ void k_final(Params P) {
  int i = blockIdx.x * blockDim.x + threadIdx.x;
  int tot = P.N * 112;
  if (i >= tot) return;
  if (i < P.N * 64) {
    int node = i >> 6, off = i & 63;
    float d = fmaxf(P.cnt[node], 1.0f);
    P.out[(size_t)node * 64 + off] = P.ssum[(size_t)node * 64 + off] / d;
  } else {
    int k = i - P.N * 64;
    int node = k / 48, off = k % 48;
    float d = fmaxf(P.cnt[node], 1.0f);
    P.out[(size_t)P.N * 64 + (size_t)node * 48 + off] =
        P.vsum[(size_t)node * 48 + off] / d;
  }
}

extern "C" void kernel_launch(void* const* d_in, const int* in_sizes, int n_in,
                              void* d_out, int out_size, void* d_ws,
                              size_t ws_size, hipStream_t stream) {
  Params P;
  P.xs = (const float*)d_in[0];
  P.xv = (const float*)d_in[1];
  const int* ei = (const int*)d_in[2];
  P.es_in = (const float*)d_in[3];
  P.ev_in = (const float*)d_in[4];
  P.rot = (const float*)d_in[5];
  P.N = in_sizes[0] / 64;
  P.E = in_sizes[2] / 2;
  P.esrc = ei;
  P.edst = ei + P.E;
  auto svp = [&](int b) {
    SvpW w;
    w.Wh = (const float*)d_in[b + 0];
    w.Ws = (const float*)d_in[b + 1];
    w.bs = (const float*)d_in[b + 2];
    w.Wmu = (const float*)d_in[b + 3];
    w.Wg = (const float*)d_in[b + 4];
    w.bg = (const float*)d_in[b + 5];
    return w;
  };
  P.tgt = svp(6);
  P.srcw = svp(12);
  P.edg = svp(18);
  P.o1 = svp(24);
  P.o2 = svp(30);
  P.v2s = (const float*)d_in[36];
  P.s2v = (const float*)d_in[37];

  float* w = (float*)d_ws;
  size_t off = 0;
  const size_t E = (size_t)P.E, N = (size_t)P.N;
  P.ws_srs = w + off; off += E * 64;
  P.ws_srv = w + off; off += E * 48;
  P.ws_es = w + off;  off += E * 64;
  P.ws_evv = w + off; off += E * 48;
  P.ws_vfs = w + off; off += E * 64;
  P.ws_sfv = w + off; off += E * 16;
  P.ws_lg = w + off;  off += E * 16;
  P.nmax = w + off;   off += N * 16;
  P.nden = w + off;   off += N * 16;
  P.cnt = w + off;    off += N;
  P.ssum = w + off;   off += N * 64;
  P.vsum = w + off;   off += N * 48;
  P.out = (float*)d_out;

  const int ntiles = (P.E + 15) / 16;
  const int blocks_t = (ntiles + WPB - 1) / WPB;

  k_init<<<(P.N * 64 + 255) / 256, 256, 0, stream>>>(P);
  k_front<<<blocks_t, 32 * WPB, 0, stream>>>(P);
  k_soft<<<(P.E * 16 + 255) / 256, 256, 0, stream>>>(P);
  k_back<<<blocks_t, 32 * WPB, 0, stream>>>(P);
  k_final<<<(P.N * 112 + 255) / 256, 256, 0, stream>>>(P);
}